// GaussianAngleDistanceLoss_72919954752051
// MI455X (gfx1250) — compile-verified
//
#include <hip/hip_runtime.h>
#include <math.h>

typedef float v4f __attribute__((ext_vector_type(4)));

#define EPSF 1.1920928955078125e-07f   // np.finfo(np.float32).eps
#define NAN_SUB 10.0f

// Compute the Gaussian-angle distance for one (A1,A2) record pair.
// a[0..5] = A[0][0], A[0][1], A[0][2], A[1][0], A[1][1], A[1][2]
__device__ __forceinline__ float gad_one(const float* __restrict__ a,
                                         const float* __restrict__ b) {
    const float a00 = a[0], a01 = a[1], a02 = a[2], a10 = a[3], a11 = a[4], a12 = a[5];
    const float b00 = b[0], b01 = b[1], b02 = b[2], b10 = b[3], b11 = b[4], b12 = b[5];

    // det of sub = A[:2,:2]; note det(cov) = det(-sub) = det(sub) for 2x2
    const float dsub1 = a00 * a11 - a01 * a10;
    const float dsub2 = b00 * b11 - b01 * b10;
    const float inv_d1 = 1.0f / dsub1;
    const float inv_d2 = 1.0f / dsub2;

    // centers: m = inv(sub) @ (-A[:2,2])
    const float m1x = (a01 * a12 - a11 * a02) * inv_d1;
    const float m1y = (a10 * a02 - a00 * a12) * inv_d1;
    const float m2x = (b01 * b12 - b11 * b02) * inv_d2;
    const float m2y = (b10 * b02 - b00 * b12) * inv_d2;

    const float det1 = fmaxf(dsub1, EPSF);
    const float det2 = fmaxf(dsub2, EPSF);

    // cov_sum = cov1 + cov2 = -(sub1 + sub2)
    const float cs00 = -(a00 + b00), cs01 = -(a01 + b01);
    const float cs10 = -(a10 + b10), cs11 = -(a11 + b11);
    const float dcs = cs00 * cs11 - cs01 * cs10;      // raw det (used in inverse)
    const float det_sum = fmaxf(dcs, EPSF);           // clamped det (used in frac)

    const float frac = 4.0f * sqrtf(det1 * det2) / det_sum;

    // inv(cov_sum) = adj / dcs
    const float idcs = 1.0f / dcs;
    const float i00 =  cs11 * idcs, i01 = -cs01 * idcs;
    const float i10 = -cs10 * idcs, i11 =  cs00 * idcs;

    // cov1, cov2
    const float c100 = -a00, c101 = -a01, c110 = -a10, c111 = -a11;
    const float c200 = -b00, c201 = -b01, c210 = -b10, c211 = -b11;

    // T = inv(cov_sum) @ cov2 ; M = cov1 @ T
    const float t00 = i00 * c200 + i01 * c210;
    const float t01 = i00 * c201 + i01 * c211;
    const float t10 = i10 * c200 + i11 * c210;
    const float t11 = i10 * c201 + i11 * c211;
    const float M00 = c100 * t00 + c101 * t10;
    const float M01 = c100 * t01 + c101 * t11;
    const float M10 = c110 * t00 + c111 * t10;
    const float M11 = c110 * t01 + c111 * t11;

    const float dx = m1x - m2x, dy = m1y - m2y;
    const float quad = dx * (M00 * dx + M01 * dy) + dy * (M10 * dx + M11 * dy);

    float earg = -0.5f * quad;
    earg = fminf(fmaxf(earg, -50.0f), 50.0f);
    const float v = frac * expf(earg);

    float dist = acosf(v);
    // nan_to_num(nan=10.0); acos output is [0,pi] or NaN, never inf
    if (dist != dist) dist = NAN_SUB;
    return dist;
}

__global__ __launch_bounds__(256)
void gad_kernel(const float* __restrict__ A1, const float* __restrict__ A2,
                float* __restrict__ out, int n) {
    const int n_chunks = (n + 3) >> 2;                 // 4 records per chunk
    const int stride = (int)(gridDim.x * blockDim.x);

    for (int c = (int)(blockIdx.x * blockDim.x + threadIdx.x); c < n_chunks; c += stride) {
        const int r0 = c << 2;

        // Prefetch next iteration's records (one iteration's working set fits
        // in the 192MB L2) -> global_prefetch_b8 (gfx1250).
        const int cn = c + stride;
        if (cn < n_chunks) {
            __builtin_prefetch(A1 + (size_t)cn * 36, 0, 0);
            __builtin_prefetch(A2 + (size_t)cn * 36, 0, 0);
        }

        if (r0 + 4 <= n) {
            // 4 records = 144B = nine aligned b128 loads per array (144*c % 16 == 0)
            const v4f* __restrict__ p1 = (const v4f*)(A1 + (size_t)r0 * 9);
            const v4f* __restrict__ p2 = (const v4f*)(A2 + (size_t)r0 * 9);
            float s1[36], s2[36];
            #pragma unroll
            for (int k = 0; k < 9; ++k) {
                const v4f u = __builtin_nontemporal_load(p1 + k);
                const v4f w = __builtin_nontemporal_load(p2 + k);
                s1[4 * k + 0] = u[0]; s1[4 * k + 1] = u[1];
                s1[4 * k + 2] = u[2]; s1[4 * k + 3] = u[3];
                s2[4 * k + 0] = w[0]; s2[4 * k + 1] = w[1];
                s2[4 * k + 2] = w[2]; s2[4 * k + 3] = w[3];
            }
            v4f o;
            o[0] = gad_one(s1 + 0,  s2 + 0);
            o[1] = gad_one(s1 + 9,  s2 + 9);
            o[2] = gad_one(s1 + 18, s2 + 18);
            o[3] = gad_one(s1 + 27, s2 + 27);
            __builtin_nontemporal_store(o, (v4f*)(out + r0));   // aligned b128 store
        } else {
            // tail (not hit for N = 4M, kept for generality)
            for (int j = 0; j < 4 && (r0 + j) < n; ++j) {
                const float* base1 = A1 + (size_t)(r0 + j) * 9;
                const float* base2 = A2 + (size_t)(r0 + j) * 9;
                float t1[6], t2[6];
                #pragma unroll
                for (int k = 0; k < 6; ++k) { t1[k] = base1[k]; t2[k] = base2[k]; }
                out[r0 + j] = gad_one(t1, t2);
            }
        }
    }
}

extern "C" void kernel_launch(void* const* d_in, const int* in_sizes, int n_in,
                              void* d_out, int out_size, void* d_ws, size_t ws_size,
                              hipStream_t stream) {
    const float* A1 = (const float*)d_in[0];
    const float* A2 = (const float*)d_in[1];
    float* out = (float*)d_out;

    const int n = in_sizes[0] / 9;                 // records of 3x3 floats
    const int n_chunks = (n + 3) / 4;
    const int TPB = 256;                           // 8 wave32 per block
    const int ITERS = 2;                           // grid-stride iterations/thread
    int blocks = (n_chunks + TPB * ITERS - 1) / (TPB * ITERS);
    if (blocks < 1) blocks = 1;

    gad_kernel<<<blocks, TPB, 0, stream>>>(A1, A2, out, n);
}